// LocalAttnReshape_58849641890017
// MI455X (gfx1250) — compile-verified
//
#include <hip/hip_runtime.h>
#include <hip/hip_bf16.h>
#include <stdint.h>

// Depth-to-space k=3 (CRD): out[b,0,3i+r,3j+s] = in[b,3r+s,i,j]
// B=32, C=9, H=W=512  ->  out (32,1,1536,1536), fp32.
// Pure data movement: 604 MB total -> ~26 us floor at 23.3 TB/s.
// Path: async global->LDS (ASYNCcnt) with NT hints, ds interleave,
//       async LDS->global with shared base + IOFFSET immediates.

#define KK     3
#define HH     512
#define WW     512
#define CHN    9                 // KK*KK
#define BATCH  32
#define OUTW   (KK*WW)           // 1536
#define OUTH   (KK*HH)           // 1536
#define TILEF  (CHN*WW)          // 4608 floats per tile (in == out size)
#define TPB    384               // 12 waves (wave32)

__global__ __launch_bounds__(TPB)
void d2s3_kernel(const float* __restrict__ in, float* __restrict__ out) {
    __shared__ float sIn[TILEF];   // input layout: [c][j]   (9 x 512)
    __shared__ float sOut[TILEF];  // output layout: [r][col] (3 x 1536)

    const int tile = blockIdx.x;       // b*H + i
    const int b = tile >> 9;           // / 512
    const int i = tile & (HH - 1);
    const int t = threadIdx.x;

    // ---- Phase 1: async global -> LDS (3 x B128 per thread, coalesced) ----
    #pragma unroll
    for (int l = 0; l < 3; ++l) {
        const int e4 = l * TPB + t;          // 0..1151 (16B chunk index)
        const int c  = e4 >> 7;              // 128 chunks per 512-float row
        const int j  = (e4 & 127) << 2;
        const float* gsrc = in + (((size_t)(b * CHN + c) * HH + i) * WW + j);
        const unsigned ldsDst = (unsigned)(uintptr_t)(&sIn[e4 << 2]);
        asm volatile("global_load_async_to_lds_b128 %0, %1, off th:TH_LOAD_NT"
                     :: "v"(ldsDst), "v"((unsigned long long)(uintptr_t)gsrc)
                     : "memory");
    }
    asm volatile("s_wait_asynccnt 0" ::: "memory");
    __syncthreads();

    // ---- Phase 2: 3-way interleave through LDS ----
    {
        const int r  = t >> 7;               // output sub-row 0..2
        const int q  = t & 127;              // 12-element group within row
        const int j0 = q << 2;
        const float4 a0 = *(const float4*)&sIn[(3 * r + 0) * WW + j0];
        const float4 a1 = *(const float4*)&sIn[(3 * r + 1) * WW + j0];
        const float4 a2 = *(const float4*)&sIn[(3 * r + 2) * WW + j0];
        // out col 12q+m: m%3 selects channel s, m/3 selects j0+m/3
        const float4 o0 = make_float4(a0.x, a1.x, a2.x, a0.y);
        const float4 o1 = make_float4(a1.y, a2.y, a0.z, a1.z);
        const float4 o2 = make_float4(a2.z, a0.w, a1.w, a2.w);
        float* dst = &sOut[r * OUTW + 12 * q];
        ((float4*)dst)[0] = o0;
        ((float4*)dst)[1] = o1;
        ((float4*)dst)[2] = o2;
    }
    __syncthreads();

    // ---- Phase 3: async LDS -> global, one base + IOFFSET immediates ----
    // INST_OFFSET is added to BOTH the global and the LDS address (ISA 08
    // async pseudocode); sOut mirrors the 3 output rows, so the per-row
    // delta (6144 B) matches on both sides.
    {
        const int col = t << 2;              // 0..1532, 384 chunks per row
        float* gdst = out + (((size_t)b * OUTH + 3 * i) * OUTW + col);
        const unsigned ldsSrc = (unsigned)(uintptr_t)(&sOut[col]);
        asm volatile(
            "global_store_async_from_lds_b128 %0, %1, off th:TH_STORE_NT\n\t"
            "global_store_async_from_lds_b128 %0, %1, off offset:6144 th:TH_STORE_NT\n\t"
            "global_store_async_from_lds_b128 %0, %1, off offset:12288 th:TH_STORE_NT"
            :: "v"((unsigned long long)(uintptr_t)gdst), "v"(ldsSrc)
            : "memory");
    }
    // s_endpgm performs an implicit wait-idle, draining ASYNCcnt.
}

extern "C" void kernel_launch(void* const* d_in, const int* in_sizes, int n_in,
                              void* d_out, int out_size, void* d_ws, size_t ws_size,
                              hipStream_t stream) {
    const float* in  = (const float*)d_in[0];   // (32,9,512,512) f32
    float*       out = (float*)d_out;           // (32,1,1536,1536) f32
    (void)in_sizes; (void)n_in; (void)out_size; (void)d_ws; (void)ws_size;
    dim3 grid(BATCH * HH);   // 16384 tiles, one (b, i) per block
    dim3 block(TPB);
    hipLaunchKernelGGL(d2s3_kernel, grid, block, 0, stream, in, out);
}